// Idx2PixelLayer_50173807952407
// MI455X (gfx1250) — compile-verified
//
#include <hip/hip_runtime.h>
#include <stdint.h>

typedef float __attribute__((ext_vector_type(4))) f4;

#define TEX_H 2048
#define TEX_W 2048
#define TEX_C 8
#define BLK   256

__global__ __launch_bounds__(BLK) void
bilerp_gather_kernel(const float* __restrict__ coords,
                     const float* __restrict__ visible,
                     const float* __restrict__ bias,
                     float* __restrict__ out,
                     int n)
{
    __shared__ float lds_coords[BLK * 2];   // 2 KB coord stage per block

    const int tid = threadIdx.x;
    const int gid = blockIdx.x * BLK + tid;

    // ---- CDNA5 async DMA: stage this block's 256 float2 coords into LDS ----
    // GVS addressing: mem = SADDR(64) + VADDR(32) + ioffset; VDST holds LDS byte addr.
    {
        // LDS byte address of this lane's 8-byte slot (low 32 bits of the flat
        // address of an LDS object are its workgroup-relative LDS offset).
        unsigned lds_byte = (unsigned)(uintptr_t)(&lds_coords[0]) + (unsigned)tid * 8u;
        // Clamp OOB lanes to a valid address (tail-safe; full blocks unaffected).
        unsigned pidx = (gid < n) ? (unsigned)gid : (unsigned)(n - 1);
        unsigned glb_byte = pidx * 8u;       // byte offset from coords base (<= 32 MB)
        asm volatile("global_load_async_to_lds_b64 %0, %1, %2"
                     :
                     : "v"(lds_byte), "v"(glb_byte), "s"(coords)
                     : "memory");
    }

    // Uniform bias load overlaps the async DMA (scalar path).
    const float b0 = bias[0];
    const float b1 = bias[1];

    asm volatile("s_wait_asynccnt 0" ::: "memory");

    if (gid >= n) return;

    // ---- per-point bilinear setup (matches reference clamping exactly) ----
    float cx = lds_coords[2 * tid + 0] - b0;   // indexes dim0 (rows, H)
    float cy = lds_coords[2 * tid + 1] - b1;   // indexes dim1 (cols, W)
    cx = fmaxf(cx, 0.0f);
    cy = fmaxf(cy, 0.0f);
    const float lx = fminf(floorf(cx), (float)(TEX_H - 2));
    const float ly = fminf(floorf(cy), (float)(TEX_W - 2));
    const float d0 = cx - lx;
    const float d1 = cy - ly;
    const int i0 = (int)lx;
    const int i1 = (int)ly;

    // tl = visible[i0, i1], tr = visible[i0+1, i1],
    // bl = visible[i0, i1+1], br = visible[i0+1, i1+1]; each 8 f32 = 32 B aligned.
    const float* base = visible + ((size_t)i0 * TEX_W + (size_t)i1) * TEX_C;

    // 8 independent 16B loads off one base (imm offsets 0,16,32,48,65536,...):
    // default RT temporal hint -> texture stays resident in the 192 MB L2.
    const f4* ptl = (const f4*)(base);
    const f4* pbl = (const f4*)(base + TEX_C);                   // +32 B
    const f4* ptr_ = (const f4*)(base + (size_t)TEX_W * TEX_C);  // +64 KB
    const f4* pbr = (const f4*)(base + (size_t)TEX_W * TEX_C + TEX_C);

    f4 tl0 = ptl[0],  tl1 = ptl[1];
    f4 bl0 = pbl[0],  bl1 = pbl[1];
    f4 tr0 = ptr_[0], tr1 = ptr_[1];
    f4 br0 = pbr[0],  br1 = pbr[1];

    // mid_bottom = br + d0*(bl-br); mid_top = tr + d0*(tl-tr);
    // interp = mid_bottom + d1*(mid_top - mid_bottom)
    f4 mb0 = br0 + d0 * (bl0 - br0);
    f4 mb1 = br1 + d0 * (bl1 - br1);
    f4 mt0 = tr0 + d0 * (tl0 - tr0);
    f4 mt1 = tr1 + d0 * (tl1 - tr1);
    f4 r0  = mb0 + d1 * (mt0 - mb0);
    f4 r1  = mb1 + d1 * (mt1 - mb1);

    // Streaming output (written once): non-temporal stores so the 128 MB
    // result stream does not evict the L2-resident texture.
    f4* po = (f4*)(out + (size_t)gid * TEX_C);
    __builtin_nontemporal_store(r0, po);
    __builtin_nontemporal_store(r1, po + 1);
}

extern "C" void kernel_launch(void* const* d_in, const int* in_sizes, int n_in,
                              void* d_out, int out_size, void* d_ws, size_t ws_size,
                              hipStream_t stream) {
    const float* coords  = (const float*)d_in[0];   // (N, 2) f32
    const float* visible = (const float*)d_in[1];   // (2048, 2048, 8) f32
    const float* bias    = (const float*)d_in[2];   // (2,) f32
    float* out = (float*)d_out;                     // (N, 8) f32

    const int n = in_sizes[0] / 2;                  // N points
    const int blocks = (n + BLK - 1) / BLK;

    bilerp_gather_kernel<<<blocks, BLK, 0, stream>>>(coords, visible, bias, out, n);
}